// InferenceLayer_1778116461329
// MI455X (gfx1250) — compile-verified
//
#include <hip/hip_runtime.h>
#include <hip/hip_bf16.h>

typedef __attribute__((ext_vector_type(2))) float v2f;
typedef __attribute__((ext_vector_type(8))) float v8f;

#define NB_CLASSES 4
#define ALPHA_ 32.0f
#define MRG_ 0.05f
#define CLW_ 0.05f
#define ZPRUNE_ 0.3f

__device__ __forceinline__ float softplusf(float x) {
    // numerically stable log1p(exp(x))
    return x > 0.f ? x + log1pf(expf(-x)) : log1pf(expf(x));
}

__device__ __forceinline__ unsigned map_f2u(float f) {
    unsigned b = __float_as_uint(f);
    return b ^ ((b & 0x80000000u) ? 0xFFFFFFFFu : 0x80000000u);
}
__device__ __forceinline__ float unmap_u2f(unsigned u) {
    unsigned b = (u & 0x80000000u) ? (u ^ 0x80000000u) : ~u;
    return __uint_as_float(b);
}

// ---------------------------------------------------------------------------
// Kernel 1: lens, zero accumulators, normalize proxies, build B-matrices
// Bmat layout: [D][16] row-major; cols 0-3 = normalized proxies, (edge only)
// col 4 = w_S, col 5 = w_E, rest zero.
// ---------------------------------------------------------------------------
__global__ __launch_bounds__(256) void setup_kernel(
    const int* __restrict__ am, const float* __restrict__ Pe,
    const float* __restrict__ Ps, const float* __restrict__ wS,
    const float* __restrict__ wE, float* __restrict__ Be,
    float* __restrict__ Bs, float* __restrict__ accums, int* __restrict__ lens,
    int Bb, int L, int D)
{
    int tid = threadIdx.x;
    if (tid < Bb) {
        int s = 0;
        for (int i = 0; i < L; ++i) s += am[tid * L + i];
        lens[tid] = s;
    }
    int nacc = 3 * 2 * Bb * NB_CLASSES + 2;  // Psim, Nsim, Pcnt, lossS, lossE
    for (int i = tid; i < nacc; i += blockDim.x) accums[i] = 0.f;

    __shared__ float pnorm[8];
    int w = tid >> 5, ln = tid & 31;
    {
        int t = w >> 2, c = w & 3;
        const float* P = (t == 0) ? Pe : Ps;
        float s = 0.f;
        for (int k = ln; k < D; k += 32) { float v = P[c * D + k]; s += v * v; }
        for (int off = 16; off >= 1; off >>= 1) s += __shfl_xor(s, off, 32);
        if (ln == 0) pnorm[w] = sqrtf(s);
    }
    __syncthreads();

    int tot = D * 16;
    for (int i = tid; i < tot; i += blockDim.x) {
        int k = i >> 4, c = i & 15;
        float ve = 0.f, vs = 0.f;
        if (c < NB_CLASSES) {
            ve = Pe[c * D + k] / fmaxf(pnorm[c], 1e-12f);
            vs = Ps[c * D + k] / fmaxf(pnorm[4 + c], 1e-12f);
        } else if (c == 4) ve = wS[k];
        else if (c == 5)   ve = wE[k];
        Be[i] = ve;
        Bs[i] = vs;
    }
}

// ---------------------------------------------------------------------------
// Kernel 2: fused streaming WMMA pass. One wave32 per 16-row tile.
// D = A(16xD) x B(Dx16) via V_WMMA_F32_16X16X4_F32 (192 steps for D=768).
// Also computes row norms, proxy-anchor exp sums, BCE partials, sigmoids.
// ---------------------------------------------------------------------------
__global__ __launch_bounds__(256) void main_kernel(
    const float* __restrict__ Xe, const float* __restrict__ Xs,
    const float* __restrict__ Be, const float* __restrict__ Bs,
    const int* __restrict__ labE, const int* __restrict__ labS,
    const int* __restrict__ tlS, const int* __restrict__ tlE,
    const float* __restrict__ bSp, const float* __restrict__ bEp,
    const int* __restrict__ lens,
    float* __restrict__ Psim, float* __restrict__ Nsim,
    float* __restrict__ Pcnt, float* __restrict__ lossAcc,
    float* __restrict__ Spred, float* __restrict__ Epred,
    int Bb, int L, int D, int tilesPerTable)
{
    int wid  = blockIdx.x * (blockDim.x >> 5) + (threadIdx.x >> 5);
    int lane = threadIdx.x & 31;
    if (wid >= 2 * tilesPerTable) return;  // wave-uniform

    int table = wid / tilesPerTable;
    int tile  = wid % tilesPerTable;
    const float* X  = (table == 0) ? Xe : Xs;
    const float* Bm = (table == 0) ? Be : Bs;
    const int*  lab = (table == 0) ? labE : labS;

    int rowsPerBatch = L * L;
    long n0   = (long)tile * 16;
    int batch = (int)(n0 / rowsPerBatch);
    int hi    = lane >> 4;       // half-wave id (K offset / M offset selector)
    int col   = lane & 15;       // A-row for this lane; D-column for this lane
    int kOff  = 2 * hi;

    const float* rowPtr = X + (n0 + (long)col) * (long)D;

    v8f acc = {0.f, 0.f, 0.f, 0.f, 0.f, 0.f, 0.f, 0.f};
    float nrm = 0.f;

    for (int k = 0; k < D; k += 4) {
        if ((k & 31) == 0 && (k + 32) < D)
            __builtin_prefetch(rowPtr + k + 32, 0, 1);  // global_prefetch_b8
        // A fragment (f32 16x4): VGPR0 = {K=k | K=k+2}, VGPR1 = {K=k+1 | K=k+3}
        float ax = rowPtr[k + kOff];
        float ay = rowPtr[k + kOff + 1];
        v2f a; a.x = ax; a.y = ay;
        nrm += ax * ax + ay * ay;
        // B fragment (f32 4x16): same half-split K layout, N striped over lanes
        v2f b;
        b.x = Bm[(k + kOff) * 16 + col];
        b.y = Bm[(k + kOff + 1) * 16 + col];
        acc = __builtin_amdgcn_wmma_f32_16x16x4_f32(
            /*neg_a=*/false, a, /*neg_b=*/false, b,
            /*c_mod=*/(short)0, acc, /*reuse_a=*/false, /*reuse_b=*/false);
    }

    // full sum-of-squares per row (row index == lane%16 for both halves)
    float nrmFull = nrm + __shfl_xor(nrm, 16, 32);
    float normS = sqrtf(nrmFull);  // lane l holds ||row l%16||

    float bS = bSp[0], bE = bEp[0];
    int lenb = lens[batch];

    float posA = 0.f, negA = 0.f, pcA = 0.f, lS = 0.f, lE = 0.f;
    for (int j = 0; j < 8; ++j) {
        int r = hi * 8 + j;            // D-matrix row held in acc[j] for this lane
        long n = n0 + r;
        int pos = (int)(n % rowsPerBatch);
        int ii = pos / L, jj = pos % L;
        float valid = (ii >= 1 && ii < lenb - 1 && jj >= 1 && jj < lenb - 1)
                          ? 1.f : 0.f;
        float nr = __shfl(normS, r, 32);   // ds_bpermute, per-lane src
        float d = acc[j];
        if (col < NB_CLASSES) {
            int lb = lab[n];
            float cosv = d / fmaxf(nr, 1e-12f);
            if (lb == col) {
                posA += valid * expf(-ALPHA_ * (cosv - MRG_));
                pcA  += valid;
            } else {
                negA += valid * expf(ALPHA_ * (cosv + MRG_));
            }
        } else if (table == 0 && col == 4) {
            int t = tlS[n];
            float wt = (t >= 0) ? 1.f : 0.f;
            float x = d + bS;
            lS += wt * (softplusf(x) - x * (float)t);
            Spred[n] = wt / (1.f + expf(-x));
        } else if (table == 0 && col == 5) {
            int t = tlE[n];
            float wt = (t >= 0) ? 1.f : 0.f;
            float x = d + bE;
            lE += wt * (softplusf(x) - x * (float)t);
            Epred[n] = wt / (1.f + expf(-x));
        }
    }

    // combine the two half-wave partials per column, then one atomic per tile
    float posT = posA + __shfl_xor(posA, 16, 32);
    float negT = negA + __shfl_xor(negA, 16, 32);
    float pcT  = pcA  + __shfl_xor(pcA, 16, 32);
    float lST  = lS   + __shfl_xor(lS, 16, 32);
    float lET  = lE   + __shfl_xor(lE, 16, 32);
    if (hi == 0) {
        if (col < NB_CLASSES) {
            int idx = (table * Bb + batch) * NB_CLASSES + col;
            atomicAdd(&Psim[idx], posT);
            atomicAdd(&Nsim[idx], negT);
            atomicAdd(&Pcnt[idx], pcT);
        } else if (table == 0 && col == 4) {
            atomicAdd(&lossAcc[0], lST);
        } else if (table == 0 && col == 5) {
            atomicAdd(&lossAcc[1], lET);
        }
    }
}

// ---------------------------------------------------------------------------
// Kernel 3: fold accumulators into the 4 scalar outputs
// ---------------------------------------------------------------------------
__global__ void finalize_kernel(const float* __restrict__ Psim,
                                const float* __restrict__ Nsim,
                                const float* __restrict__ Pcnt,
                                const float* __restrict__ lossAcc,
                                float* __restrict__ out, int Bb, int L)
{
    if (threadIdx.x != 0 || blockIdx.x != 0) return;
    for (int t = 0; t < 2; ++t) {
        float cl = 0.f;
        for (int b = 0; b < Bb; ++b) {
            int nv = 0;
            for (int c = 0; c < NB_CLASSES; ++c)
                if (Pcnt[(t * Bb + b) * NB_CLASSES + c] > 0.f) nv++;
            float nvf = (float)(nv < 1 ? 1 : nv);
            float s = 0.f;
            for (int c = 0; c < NB_CLASSES; ++c) {
                s += log1pf(Psim[(t * Bb + b) * NB_CLASSES + c]) / nvf;
                s += log1pf(Nsim[(t * Bb + b) * NB_CLASSES + c]) *
                     (1.f / (float)NB_CLASSES);
            }
            cl += s;
        }
        out[t] = CLW_ * cl / (float)Bb;
    }
    float denom = (float)Bb * (float)L * (float)L;
    out[2] = lossAcc[0] / denom;
    out[3] = lossAcc[1] / denom;
}

// ---------------------------------------------------------------------------
// Kernel 4: exact k-th-largest per (table,batch) via MSB-first radix select
// ---------------------------------------------------------------------------
__global__ __launch_bounds__(256) void select_kernel(
    const float* __restrict__ Spred, const float* __restrict__ Epred,
    const int* __restrict__ lens, float* __restrict__ thresh, int Bb, int L)
{
    int sel = blockIdx.x;          // 0..Bb-1 = S, Bb..2Bb-1 = E
    int b = sel % Bb;
    const float* vals = ((sel < Bb) ? Spred : Epred) + (long)b * L * L;
    int N = L * L;

    int ml = lens[b] - 2;
    int length = (int)((float)ml * ZPRUNE_);
    if (length < 10) length = 10;
    int cap = ml * ml;
    if (length > cap) length = cap;

    __shared__ unsigned hist[256];
    __shared__ unsigned sh_prefix;
    __shared__ int sh_k;
    if (threadIdx.x == 0) { sh_prefix = 0u; sh_k = length; }
    __syncthreads();

    for (int pass = 0; pass < 4; ++pass) {
        int shift = 24 - pass * 8;
        unsigned maskHigh = (pass == 0) ? 0u : (0xFFFFFFFFu << (shift + 8));
        hist[threadIdx.x] = 0u;
        __syncthreads();
        unsigned prefix = sh_prefix;
        for (int i = threadIdx.x; i < N; i += 256) {
            unsigned u = map_f2u(vals[i]);
            if ((u & maskHigh) == prefix)
                atomicAdd(&hist[(u >> shift) & 0xFFu], 1u);
        }
        __syncthreads();
        if (threadIdx.x == 0) {
            int k = sh_k;
            unsigned cum = 0;
            int chosen = 0;
            for (int v = 255; v >= 0; --v) {
                unsigned c = hist[v];
                if (cum + c >= (unsigned)k) { chosen = v; sh_k = k - (int)cum; break; }
                cum += c;
            }
            sh_prefix = prefix | ((unsigned)chosen << shift);
        }
        __syncthreads();
    }
    if (threadIdx.x == 0) thresh[sel] = unmap_u2f(sh_prefix);
}

// ---------------------------------------------------------------------------
// Kernel 5: pred = (val >= thresh[b]) as 1.0/0.0 into d_out
// ---------------------------------------------------------------------------
__global__ __launch_bounds__(256) void compare_kernel(
    const float* __restrict__ Spred, const float* __restrict__ Epred,
    const float* __restrict__ thresh, float* __restrict__ out, int Bb, int L)
{
    long rows = (long)Bb * L * L;
    long idx = (long)blockIdx.x * blockDim.x + threadIdx.x;
    if (idx >= 2 * rows) return;
    int LL = L * L;
    if (idx < rows) {
        int b = (int)(idx / LL);
        out[4 + idx] = (Spred[idx] >= thresh[b]) ? 1.f : 0.f;
    } else {
        long e = idx - rows;
        int b = (int)(e / LL);
        out[4 + rows + e] = (Epred[e] >= thresh[Bb + b]) ? 1.f : 0.f;
    }
}

// ---------------------------------------------------------------------------
extern "C" void kernel_launch(void* const* d_in, const int* in_sizes, int n_in,
                              void* d_out, int out_size, void* d_ws,
                              size_t ws_size, hipStream_t stream)
{
    const float* Xe  = (const float*)d_in[0];   // table_edge
    const float* Xs  = (const float*)d_in[1];   // table_senti
    const int*   am  = (const int*)d_in[2];     // attention_mask
    const int*   tlS = (const int*)d_in[3];     // table_labels_S
    const int*   tlE = (const int*)d_in[4];     // table_labels_E
    const int*   labE = (const int*)d_in[5];    // labels_edge
    const int*   labS = (const int*)d_in[6];    // labels_senti
    const float* Pe  = (const float*)d_in[7];   // proxies_edge
    const float* Ps  = (const float*)d_in[8];   // proxies_senti
    const float* wS  = (const float*)d_in[9];
    const float* bS  = (const float*)d_in[10];
    const float* wE  = (const float*)d_in[11];
    const float* bE  = (const float*)d_in[12];

    int D   = in_sizes[9];               // 768
    int BL  = in_sizes[2];               // B*L
    int BLL = in_sizes[3];               // B*L*L
    int L   = BLL / BL;                  // 128
    int Bb  = BL / L;                    // 4
    long rows = (long)Bb * L * L;        // 65536
    int tiles = (int)(rows / 16);        // 4096

    // workspace layout (floats)
    float* ws     = (float*)d_ws;
    float* Be     = ws;
    float* Bs     = Be + (size_t)D * 16;
    float* Spred  = Bs + (size_t)D * 16;
    float* Epred  = Spred + rows;
    float* Psim   = Epred + rows;                 // start of accumulators
    float* Nsim   = Psim + 2 * Bb * NB_CLASSES;
    float* Pcnt   = Nsim + 2 * Bb * NB_CLASSES;
    float* lossA  = Pcnt + 2 * Bb * NB_CLASSES;
    float* thresh = lossA + 2;
    int*   lens   = (int*)(thresh + 2 * Bb);

    setup_kernel<<<1, 256, 0, stream>>>(am, Pe, Ps, wS, wE, Be, Bs,
                                        Psim, lens, Bb, L, D);

    int wavesTotal = 2 * tiles;
    int nBlocks = (wavesTotal + 7) / 8;  // 8 waves per 256-thread block
    main_kernel<<<nBlocks, 256, 0, stream>>>(
        Xe, Xs, Be, Bs, labE, labS, tlS, tlE, bS, bE, lens,
        Psim, Nsim, Pcnt, lossA, Spred, Epred, Bb, L, D, tiles);

    finalize_kernel<<<1, 32, 0, stream>>>(Psim, Nsim, Pcnt, lossA,
                                          (float*)d_out, Bb, L);

    select_kernel<<<2 * Bb, 256, 0, stream>>>(Spred, Epred, lens, thresh, Bb, L);

    long n2 = 2 * rows;
    compare_kernel<<<(int)((n2 + 255) / 256), 256, 0, stream>>>(
        Spred, Epred, thresh, (float*)d_out, Bb, L);
}